// Attention_Model_87522843558695
// MI455X (gfx1250) — compile-verified
//
#include <hip/hip_runtime.h>

// CDNA5 / gfx1250 fused kernel for the "attention + LSTM-cell" reference.
// Math reduction: the scan is embarrassingly parallel because the recurrent
// state and bd cancel inside softmax(axis=t), and h0=c0=0 kill the Wr and zf
// terms. Per batch:
//   S = X_b @ Wd[:128]; A = softmax_t(S); CTX = A^T @ X_b;
//   Z = CTX @ Wk (+bl);  out = sigmoid(zo)*tanh(sigmoid(zi)*tanh(zc))
// GEMMs run on V_WMMA_F32_16X16X4_F32 (exact f32; problem is latency-bound at
// ~0.7 GFLOP / ~4.3MB, so low-precision WMMA buys nothing). Staging of X_b and
// Wd into LDS uses the gfx1250 async global->LDS path (ASYNCcnt) to skip the
// VGPR round-trip.

typedef float v2f __attribute__((ext_vector_type(2)));
typedef float v8f __attribute__((ext_vector_type(8)));

#define T_   128
#define F_   128
#define U_   128
#define LDA  132   // padded LDS row stride in floats: 528B rows (16B aligned, bank-spread)

__device__ __forceinline__ float sigmoid_f(float x) {
    return 1.0f / (1.0f + __expf(-x));
}

// Async copy of 16 bytes (one float4) global -> LDS, per lane.
// ISA: GLOBAL_LOAD_ASYNC_TO_LDS_B128  vdst(LDS byte addr), vaddr(64b), saddr=off
// Tracked by ASYNCcnt; must s_wait_asynccnt before consuming / barrier-signal.
__device__ __forceinline__ void async_cp16(float* lds_dst, const float* gsrc) {
    const unsigned lds_off = (unsigned)(reinterpret_cast<uintptr_t>(lds_dst));
    asm volatile("global_load_async_to_lds_b128 %0, %1, off"
                 :: "v"(lds_off), "v"(gsrc)
                 : "memory");
}

__global__ __launch_bounds__(256)
void attn_lstm_fused(const float* __restrict__ X,    // (B,T,F)
                     const float* __restrict__ Wd,   // (F+U, T), rows 0..F-1 used
                     const float* __restrict__ Wk,   // (F, 4U)
                     const float* __restrict__ bl,   // (4U)
                     float*       __restrict__ out)  // (B,T,U)
{
    extern __shared__ float smem[];
    float* sX = smem;                 // [T_][LDA]  X_b
    float* sW = smem + T_ * LDA;      // [F_][LDA]  Wd (phase 1), CTX (phase 2+)
    float* sS = smem + 2 * T_ * LDA;  // [T_][LDA]  S, then A (in-place softmax)

    const int b    = blockIdx.x;
    const int tid  = threadIdx.x;
    const int wave = tid >> 5;        // 0..7 (wave32)
    const int lane = tid & 31;
    const int hh   = lane >> 4;       // half-wave select (0/1)
    const int l    = lane & 15;       // lane-in-half

    // ------- stage X_b and Wd[:F] into LDS via async global->LDS B128 -------
    const float* Xb = X + (size_t)b * T_ * F_;
    for (int idx = tid; idx < T_ * (F_ / 4); idx += 256) {
        const int r  = idx >> 5;           // F_/4 == 32 float4 per row
        const int c4 = (idx & 31) << 2;
        async_cp16(sX + r * LDA + c4, Xb + r * F_ + c4);
    }
    for (int idx = tid; idx < F_ * (T_ / 4); idx += 256) {
        const int r  = idx >> 5;
        const int c4 = (idx & 31) << 2;
        async_cp16(sW + r * LDA + c4, Wd + r * T_ + c4);
    }
    asm volatile("s_wait_asynccnt 0x0" ::: "memory");
    __syncthreads();

    // ---------------- phase 1: S = X_b @ WdX  (each wave: tile row mt=wave) --
    {
        const int mt = wave;
        for (int nt = 0; nt < 8; ++nt) {
            v8f acc = {};
            #pragma unroll 4
            for (int k0 = 0; k0 < F_; k0 += 4) {
                const int ka = k0 + 2 * hh;
                v2f a, bf;
                a.x  = sX[(mt * 16 + l) * LDA + ka];
                a.y  = sX[(mt * 16 + l) * LDA + ka + 1];
                bf.x = sW[ka * LDA + nt * 16 + l];
                bf.y = sW[(ka + 1) * LDA + nt * 16 + l];
                acc = __builtin_amdgcn_wmma_f32_16x16x4_f32(
                          false, a, false, bf, (short)0, acc, false, false);
            }
            #pragma unroll
            for (int v = 0; v < 8; ++v)
                sS[(mt * 16 + v + hh * 8) * LDA + nt * 16 + l] = acc[v];
        }
    }
    __syncthreads();

    // ---------------- softmax over t (down each column j) --------------------
    if (tid < T_) {
        const int j = tid;
        float m = -3.0e38f;
        for (int t = 0; t < T_; ++t) m = fmaxf(m, sS[t * LDA + j]);
        float s = 0.0f;
        for (int t = 0; t < T_; ++t) {
            const float e = __expf(sS[t * LDA + j] - m);
            sS[t * LDA + j] = e;
            s += e;
        }
        const float inv = 1.0f / s;
        for (int t = 0; t < T_; ++t) sS[t * LDA + j] *= inv;
    }
    __syncthreads();

    // ---------------- phase 2: CTX = A^T @ X_b  (overwrite sW) ---------------
    {
        const int it = wave;
        for (int ft = 0; ft < 8; ++ft) {
            v8f acc = {};
            #pragma unroll 4
            for (int k0 = 0; k0 < T_; k0 += 4) {
                const int ka = k0 + 2 * hh;
                v2f a, bf;
                // A^T: element [i][t] = A[t][i]
                a.x  = sS[ka * LDA + it * 16 + l];
                a.y  = sS[(ka + 1) * LDA + it * 16 + l];
                bf.x = sX[ka * LDA + ft * 16 + l];
                bf.y = sX[(ka + 1) * LDA + ft * 16 + l];
                acc = __builtin_amdgcn_wmma_f32_16x16x4_f32(
                          false, a, false, bf, (short)0, acc, false, false);
            }
            #pragma unroll
            for (int v = 0; v < 8; ++v)
                sW[(it * 16 + v + hh * 8) * LDA + ft * 16 + l] = acc[v];
        }
    }
    __syncthreads();

    // ---------------- phase 3: Z = CTX @ Wk, fused gates, store --------------
    // Only quadrants q=0 (zi), q=2 (zc), q=3 (zo) matter (c0 == 0 kills zf).
    {
        const int it = wave;
        float* outb = out + (size_t)b * T_ * U_;
        for (int ut = 0; ut < 8; ++ut) {
            v8f ci = {}, cc = {}, co = {};
            const int col = ut * 16 + l;
            #pragma unroll 2
            for (int k0 = 0; k0 < F_; k0 += 4) {
                const int ka = k0 + 2 * hh;
                v2f a, b0, b2, b3;
                a.x = sW[(it * 16 + l) * LDA + ka];
                a.y = sW[(it * 16 + l) * LDA + ka + 1];
                b0.x = Wk[(size_t)ka * 512 + col];
                b0.y = Wk[(size_t)(ka + 1) * 512 + col];
                b2.x = Wk[(size_t)ka * 512 + 256 + col];
                b2.y = Wk[(size_t)(ka + 1) * 512 + 256 + col];
                b3.x = Wk[(size_t)ka * 512 + 384 + col];
                b3.y = Wk[(size_t)(ka + 1) * 512 + 384 + col];
                ci = __builtin_amdgcn_wmma_f32_16x16x4_f32(
                         false, a, false, b0, (short)0, ci, false, false);
                cc = __builtin_amdgcn_wmma_f32_16x16x4_f32(
                         false, a, false, b2, (short)0, cc, false, false);
                co = __builtin_amdgcn_wmma_f32_16x16x4_f32(
                         false, a, false, b3, (short)0, co, false, false);
            }
            const int u  = ut * 16 + l;
            const float bi = bl[u];
            const float bc = bl[256 + u];
            const float bo = bl[384 + u];
            #pragma unroll
            for (int v = 0; v < 8; ++v) {
                const int row = it * 16 + v + hh * 8;
                const float zi = ci[v] + bi;
                const float zc = cc[v] + bc;
                const float zo = co[v] + bo;
                const float cv = sigmoid_f(zi) * tanhf(zc);
                outb[row * U_ + u] = sigmoid_f(zo) * tanhf(cv);
            }
        }
    }
}

extern "C" void kernel_launch(void* const* d_in, const int* in_sizes, int n_in,
                              void* d_out, int out_size, void* d_ws, size_t ws_size,
                              hipStream_t stream) {
    (void)in_sizes; (void)n_in; (void)out_size; (void)d_ws; (void)ws_size;
    const float* X  = (const float*)d_in[0];  // (32,128,128)
    const float* Wd = (const float*)d_in[1];  // (256,128), rows 0..127 used
    // d_in[2] = bd  : cancels inside softmax over t  -> unused
    const float* Wk = (const float*)d_in[3];  // (128,512)
    // d_in[4] = Wr  : multiplied by h0 == 0          -> unused
    const float* bl = (const float*)d_in[5];  // (512)
    float* out = (float*)d_out;               // (32,128,128)

    const size_t shmem = (size_t)3 * T_ * LDA * sizeof(float);  // ~198 KB < 320 KB/WGP
    attn_lstm_fused<<<dim3(32), dim3(256), shmem, stream>>>(X, Wd, Wk, bl, out);
}